// MMAKernelLayer_46583215292931
// MI455X (gfx1250) — compile-verified
//
#include <hip/hip_runtime.h>

// CDNA5 / gfx1250, wave32. WMMA f32 16x16x4 used for the n-reduction:
// out[m] = sum_n K[n,m] * w2[n], accumulated 4 n's per v_wmma_f32_16x16x4_f32.

typedef float v2f __attribute__((ext_vector_type(2)));
typedef float v8f __attribute__((ext_vector_type(8)));

#define SAMP 4
#define NPTS 256
#define NB   16
#define ND   16
#define RES  64
#define MGRID (RES * RES)
// exp(-0.5*(x/0.05)^2) = exp(-200*x^2)
#define NEG_HALF_INV_BW2 (-200.0f)

// ---------------------------------------------------------------------------
// Kernel 1: interleaving weights  w2[s*N+n] = (max_{b,d} min_c relu(D-B))^2
// S*N = 1024 threads total; each loads 16 B-corners + 16 D-corners (float2).
// ---------------------------------------------------------------------------
__global__ void __launch_bounds__(NPTS)
pd_weights_kernel(const float* __restrict__ Bs,
                  const float* __restrict__ Ds,
                  float* __restrict__ w2) {
  const int t = blockIdx.x * blockDim.x + threadIdx.x;  // (s*N + n)
  if (t >= SAMP * NPTS) return;
  const float2* Bp = (const float2*)(Bs + (size_t)t * NB * 2);
  const float2* Dp = (const float2*)(Ds + (size_t)t * ND * 2);
  float2 bc[NB], dc[ND];
#pragma unroll
  for (int i = 0; i < NB; ++i) bc[i] = Bp[i];
#pragma unroll
  for (int i = 0; i < ND; ++i) dc[i] = Dp[i];
  float w = 0.0f;  // pd >= 0 always, so 0 == reference's -inf + isinf guard
#pragma unroll 4
  for (int d = 0; d < ND; ++d) {
#pragma unroll
    for (int b = 0; b < NB; ++b) {
      const float px = fmaxf(dc[d].x - bc[b].x, 0.0f);
      const float py = fmaxf(dc[d].y - bc[b].y, 0.0f);
      w = fmaxf(w, fminf(px, py));
    }
  }
  w2[t] = w * w;  // P == 2.0
}

// ---------------------------------------------------------------------------
// Kernel 2: one wave32 per (sample, 16-grid-point tile).
//   A (16x4 f32, 2 VGPRs): w2[n_chunk] broadcast over the 16 rows.
//     lanes 0-15: {k=0,k=1}; lanes 16-31: {k=2,k=3}   (ISA 32-bit A layout)
//   B (4x16 f32, 2 VGPRs): K[n_chunk, m_tile] computed in-register.
//     lane<16 holds rows k=0,1 for col m=lane; lane>=16 rows k=2,3 for m=lane-16
//   C/D (16x16 f32, 8 VGPRs): every row equals out[m_tile]; lanes 0-15 store
//     VGPR0 (row 0, col = lane).
// ---------------------------------------------------------------------------
__global__ void __launch_bounds__(32)
kde_wmma_kernel(const float* __restrict__ Bs,
                const float* __restrict__ Ds,
                const float* __restrict__ w2,
                float* __restrict__ out) {
  const int wave = blockIdx.x;            // 0 .. SAMP*(MGRID/16)-1
  const int s    = wave >> 8;             // wave / 256
  const int m0   = (wave & 255) << 4;     // tile base grid point
  const int lane = threadIdx.x;           // 0..31
  const int half = lane >> 4;             // 0: k={0,1}, 1: k={2,3}
  const int m    = m0 + (lane & 15);      // this lane's grid point (col)

  const float inv = 1.0f / (float)(RES - 1);
  const float x0  = (float)(m >> 6) * inv;  // g0[m / 64]
  const float x1  = (float)(m & 63) * inv;  // g1[m % 64]

  const float*  w2s = w2 + s * NPTS;
  const float2* Bp  = (const float2*)(Bs + (size_t)s * NPTS * NB * 2);
  const float2* Dp  = (const float2*)(Ds + (size_t)s * NPTS * ND * 2);

  v8f c = {0.f, 0.f, 0.f, 0.f, 0.f, 0.f, 0.f, 0.f};

  for (int nb = 0; nb < NPTS; nb += 4) {
    const int n0 = nb + 2 * half;

    v2f a, b;
    a.x = w2s[n0];
    a.y = w2s[n0 + 1];

#pragma unroll
    for (int u = 0; u < 2; ++u) {
      const int n = n0 + u;
      const float2* bcp = Bp + n * NB;
      const float2* dcp = Dp + n * ND;
      float bd = INFINITY;
      float dd = INFINITY;
#pragma unroll
      for (int k = 0; k < NB; ++k) {
        const float2 bc = bcp[k];
        bd = fminf(bd, fmaxf(bc.x - x0, bc.y - x1));
      }
#pragma unroll
      for (int k = 0; k < ND; ++k) {
        const float2 dc = dcp[k];
        dd = fminf(dd, fmaxf(x0 - dc.x, x1 - dc.y));
      }
      const float xk = fmaxf(fmaxf(bd, dd), 0.0f);
      const float Kv = __expf(NEG_HALF_INV_BW2 * xk * xk);
      if (u == 0) b.x = Kv; else b.y = Kv;
    }

    // D = A x B + C  (8 args: neg_a, A, neg_b, B, c_mod, C, reuse_a, reuse_b)
    c = __builtin_amdgcn_wmma_f32_16x16x4_f32(
        false, a, false, b, (short)0, c, false, false);
  }

  if (lane < 16) out[(size_t)s * MGRID + m0 + lane] = c[0];
}

extern "C" void kernel_launch(void* const* d_in, const int* in_sizes, int n_in,
                              void* d_out, int out_size, void* d_ws, size_t ws_size,
                              hipStream_t stream) {
  const float* Bs = (const float*)d_in[0];  // [4,256,16,2] f32
  const float* Ds = (const float*)d_in[1];  // [4,256,16,2] f32
  float* w2  = (float*)d_ws;                // 4KB scratch: w^2 per (s,n)
  float* out = (float*)d_out;               // [4, 4096] f32

  pd_weights_kernel<<<SAMP, NPTS, 0, stream>>>(Bs, Ds, w2);
  kde_wmma_kernel<<<SAMP * (MGRID / 16), 32, 0, stream>>>(Bs, Ds, w2, out);
}